// GE2E_66108136620547
// MI455X (gfx1250) — compile-verified
//
#include <hip/hip_runtime.h>
#include <math.h>

typedef __attribute__((ext_vector_type(2))) float v2f;
typedef __attribute__((ext_vector_type(4))) float v4f;
typedef __attribute__((ext_vector_type(8))) float v8f;
typedef __attribute__((ext_vector_type(4))) int   v4i;

#define N_SPK 1024
#define M_UTT 32
#define D_DIM 256
#define NM    (N_SPK * M_UTT)   // 32768
#define EPSV  1e-6f

// workspace layout (float offsets)
#define WS_CENT  0
#define WS_CNORM (WS_CENT + N_SPK * D_DIM)   // 262144
#define WS_XNORM (WS_CNORM + N_SPK)          // 263168
#define WS_DSIM  (WS_XNORM + NM)             // 295936
#define WS_PART  (WS_DSIM + NM)              // 328704
#define NBLK_B   (NM / 64)                   // 512

#define ROWLEN 260   // padded LDS row stride (floats) to break bank conflicts

#if __has_builtin(__builtin_amdgcn_global_load_async_to_lds_b128)
#define HAVE_ASYNC_LDS 1
#else
#define HAVE_ASYNC_LDS 0
#endif

// pointer-to-v4i in global (AS1) / LDS (AS3) address spaces
typedef __attribute__((address_space(1))) v4i* gas_v4i_ptr;
typedef __attribute__((address_space(3))) v4i* las_v4i_ptr;

// async global->LDS b128 copy (16B per lane), tracked by ASYNCcnt
__device__ __forceinline__ void copy16_to_lds(const float* gsrc, float* ldst) {
#if HAVE_ASYNC_LDS
    __builtin_amdgcn_global_load_async_to_lds_b128(
        (gas_v4i_ptr)(void*)gsrc,
        (las_v4i_ptr)(void*)ldst,
        /*offset=*/0, /*cpol=*/0);
#else
    *(v4f*)ldst = *(const v4f*)gsrc;   // sync fallback (load + ds_store)
#endif
}

__device__ __forceinline__ void async_copy_wait() {
#if HAVE_ASYNC_LDS
#if __has_builtin(__builtin_amdgcn_s_wait_asynccnt)
    __builtin_amdgcn_s_wait_asynccnt(0);
#else
    asm volatile("s_wait_asynccnt 0x0" ::: "memory");
#endif
#endif
    // sync fallback needs nothing: __syncthreads() fences DS stores
}

// ---------------------------------------------------------------------------
// Kernel A: per-speaker centroids + norms + modified-centroid diagonal sims
// grid = N_SPK blocks, 256 threads (thread == feature dim d)
// ---------------------------------------------------------------------------
__global__ void __launch_bounds__(256) ge2e_pre(const float* __restrict__ x,
                                                float* __restrict__ ws) {
    __shared__ float s_sum[D_DIM];
    __shared__ float s_red[17];

    const int n    = blockIdx.x;
    const int d    = threadIdx.x;
    const int lane = d & 31;
    const int wid  = d >> 5;

    const float* xn = x + (size_t)n * M_UTT * D_DIM;

    // pass 1: sum over utterances for this dim
    float sum = 0.f;
    #pragma unroll
    for (int m = 0; m < M_UTT; ++m) sum += xn[m * D_DIM + d];
    s_sum[d] = sum;

    const float c = sum * (1.0f / M_UTT);
    ws[WS_CENT + n * D_DIM + d] = c;

    float cn2 = c * c;      // -> ||cent||^2
    float S2  = sum * sum;  // -> ||sum_x||^2
    #pragma unroll
    for (int off = 16; off; off >>= 1) {
        cn2 += __shfl_xor(cn2, off, 32);
        S2  += __shfl_xor(S2,  off, 32);
    }
    if (lane == 0) { s_red[wid] = cn2; s_red[8 + wid] = S2; }
    __syncthreads();
    if (d == 0) {
        float a = 0.f, s2t = 0.f;
        for (int i = 0; i < 8; ++i) { a += s_red[i]; s2t += s_red[8 + i]; }
        ws[WS_CNORM + n] = sqrtf(a);
        s_red[16] = s2t;
    }
    __syncthreads();
    const float S2t = s_red[16];

    // pass 2: per-utterance ||x||, dot(x, sum_x) -> diag_sim via identities
    for (int i = 0; i < 4; ++i) {
        const int m = wid * 4 + i;
        float xn2 = 0.f, dxs = 0.f;
        #pragma unroll
        for (int j = 0; j < 8; ++j) {
            const int dd = lane + 32 * j;
            const float xv = xn[m * D_DIM + dd];
            xn2 += xv * xv;
            dxs += xv * s_sum[dd];
        }
        #pragma unroll
        for (int off = 16; off; off >>= 1) {
            xn2 += __shfl_xor(xn2, off, 32);
            dxs += __shfl_xor(dxs, off, 32);
        }
        if (lane == 0) {
            const float xnorm = sqrtf(xn2);
            const float dmod  = (dxs - xn2) * (1.0f / (M_UTT - 1));
            float mn2 = fmaxf(S2t - 2.0f * dxs + xn2, 0.0f) *
                        (1.0f / ((M_UTT - 1) * (M_UTT - 1)));
            const float modn = sqrtf(mn2);
            const float dsim = dmod / fmaxf(xnorm * modn, EPSV);
            ws[WS_XNORM + n * M_UTT + m] = xnorm;
            ws[WS_DSIM  + n * M_UTT + m] = dsim;
        }
    }
}

// ---------------------------------------------------------------------------
// Kernel B: fused (X * cent^T) GEMM via V_WMMA_F32_16X16X4_F32 + online LSE,
// with double-buffered async global->LDS staging of centroid tiles.
// grid = NM/64 blocks, 128 threads (4 waves, 16 rows per wave)
// ---------------------------------------------------------------------------
__global__ void __launch_bounds__(128) ge2e_main(const float* __restrict__ x,
                                                 const float* __restrict__ wp,
                                                 const float* __restrict__ bp,
                                                 const float* __restrict__ ws,
                                                 float* __restrict__ partials) {
    __shared__ float s_cent[2][16 * ROWLEN];
    __shared__ float s_part[8];

    const int tid  = threadIdx.x;
    const int wave = tid >> 5;     // 0..3
    const int lane = tid & 31;
    const int hi   = lane >> 4;    // half-wave select
    const int nl   = lane & 15;

    const int row_base = blockIdx.x * 64 + wave * 16;  // this wave's 16 rows

    const float* cent    = ws + WS_CENT;
    const float* cnorm_g = ws + WS_CNORM;
    const float* xnorm_g = ws + WS_XNORM;
    const float* dsim_g  = ws + WS_DSIM;

    const float w = wp[0];
    const float b = bp[0];

    // stage one 16x256 centroid tile into LDS buffer (8 x b128 per thread)
    auto stage = [&](int ct, int buf) {
        const float* src = cent + (size_t)ct * 16 * D_DIM;
        #pragma unroll
        for (int i = 0; i < 8; ++i) {
            const int idx = tid + i * 128;     // 0..1023 float4's
            const int r   = idx >> 6;
            const int c4  = idx & 63;
            copy16_to_lds(src + r * D_DIM + c4 * 4,
                          &s_cent[buf][r * ROWLEN + c4 * 4]);
        }
    };

    stage(0, 0);   // prologue: tile 0 in flight while we preload A

    // Preload A fragments (16x4 f32 layout: lane = row M, K = 4t + 2*hi + {0,1})
    v2f a[64];
    {
        const v2f* ap = (const v2f*)(x + (size_t)(row_base + nl) * D_DIM + 2 * hi);
        #pragma unroll
        for (int t = 0; t < 64; ++t) a[t] = ap[2 * t];
    }

    // Per-row constants (C fragment row = v + 8*hi)
    float xn[8], dsim[8];
    #pragma unroll
    for (int v = 0; v < 8; ++v) {
        const int grow = row_base + v + 8 * hi;
        xn[v]   = xnorm_g[grow];
        dsim[v] = dsim_g[grow];
    }
    const int n_tile   = row_base >> 5;  // same speaker for all 16 rows
    const int ct_diag  = n_tile >> 4;
    const int col_diag = n_tile & 15;

    float rm[8], rs[8];
    #pragma unroll
    for (int v = 0; v < 8; ++v) { rm[v] = -1e30f; rs[v] = 0.f; }

    for (int ct = 0; ct < 64; ++ct) {
        const int buf = ct & 1;
        async_copy_wait();     // my staged loads for `buf` are in LDS
        __syncthreads();       // everyone's are; prior reads of buf^1 retired
        if (ct + 1 < 64) stage(ct + 1, buf ^ 1);   // overlap with WMMA below

        v8f acc = {};
        {
            const v2f* bp2 = (const v2f*)&s_cent[buf][nl * ROWLEN + 2 * hi];
            #pragma unroll
            for (int t = 0; t < 64; ++t) {
                const v2f bf = bp2[2 * t];
                acc = __builtin_amdgcn_wmma_f32_16x16x4_f32(
                        false, a[t], false, bf, (short)0, acc, false, false);
            }
        }

        const float cnv = cnorm_g[ct * 16 + nl];   // L2-hot, 4KB total
        const bool diag_here = (ct == ct_diag) && (nl == col_diag);
        #pragma unroll
        for (int v = 0; v < 8; ++v) {
            const float sim = acc[v] / fmaxf(xn[v] * cnv, EPSV);
            float cval = w * sim + b;
            if (diag_here) cval = w * dsim[v] + b;  // splice modified-centroid sim
            const float mn = fmaxf(rm[v], cval);
            rs[v] = rs[v] * __expf(rm[v] - mn) + __expf(cval - mn);
            rm[v] = mn;
        }
    }

    // merge (max, sumexp) across the 16 lanes holding each row
    float wsum = 0.f;
    #pragma unroll
    for (int v = 0; v < 8; ++v) {
        float m = rm[v], s = rs[v];
        #pragma unroll
        for (int off = 1; off < 16; off <<= 1) {
            const float m2 = __shfl_xor(m, off, 32);
            const float s2 = __shfl_xor(s, off, 32);
            const float mm = fmaxf(m, m2);
            s = s * __expf(m - mm) + s2 * __expf(m2 - mm);
            m = mm;
        }
        const float lse = m + __logf(s);
        wsum += lse - (w * dsim[v] + b);   // -diag_calc + logsumexp
    }

    if (nl == 0) s_part[wave * 2 + hi] = wsum;
    __syncthreads();
    if (tid == 0) {
        float t = 0.f;
        for (int i = 0; i < 8; ++i) t += s_part[i];   // fixed order: deterministic
        partials[blockIdx.x] = t;
    }
}

// ---------------------------------------------------------------------------
// Kernel C: deterministic fixed-order reduction of block partials
// ---------------------------------------------------------------------------
__global__ void __launch_bounds__(256) ge2e_final(const float* __restrict__ partials,
                                                  float* __restrict__ out) {
    __shared__ float s[256];
    const int t = threadIdx.x;
    s[t] = partials[t] + partials[t + 256];
    __syncthreads();
    for (int off = 128; off; off >>= 1) {
        if (t < off) s[t] += s[t + off];
        __syncthreads();
    }
    if (t == 0) out[0] = s[0];
}

extern "C" void kernel_launch(void* const* d_in, const int* in_sizes, int n_in,
                              void* d_out, int out_size, void* d_ws, size_t ws_size,
                              hipStream_t stream) {
    const float* x = (const float*)d_in[0];
    const float* w = (const float*)d_in[1];
    const float* b = (const float*)d_in[2];
    float* ws  = (float*)d_ws;
    float* out = (float*)d_out;

    ge2e_pre  <<<N_SPK, 256, 0, stream>>>(x, ws);
    ge2e_main <<<NBLK_B, 128, 0, stream>>>(x, w, b, ws, ws + WS_PART);
    ge2e_final<<<1, 256, 0, stream>>>(ws + WS_PART, out);
}